// FalconAttention_24266565223034
// MI455X (gfx1250) — compile-verified
//
#include <hip/hip_runtime.h>
#include <hip/hip_bf16.h>
#include <math.h>

// ---------------------------------------------------------------------------
// Falcon decode attention for MI455X (gfx1250, wave32).
// fp32 end-to-end using V_WMMA_F32_16X16X4_F32 for all matrix math.
// ---------------------------------------------------------------------------

typedef float v2f __attribute__((ext_vector_type(2)));
typedef float v8f __attribute__((ext_vector_type(8)));

#define BATCH   64
#define HID     4096
#define NHEAD   32
#define HD      128
#define QKV_N   4352        // (H+2)*D
#define MAXBLK  256
#define BLKSZ   16
#define NEGINF  (-1e30f)

// ---------------------------------------------------------------------------
// GEMM: C[64 x N] = A[64 x 4096] @ W[4096 x N], fp32, one wave per 16x16 tile.
// Fragment layouts per CDNA5 ISA 7.12.2 (f32 16x16x4):
//   A 16x4 : lane(0..15|16..31) -> row M=lane&15 ; frag.x = K=2*hi, frag.y = K=2*hi+1
//   B 4x16 : lane -> col N=lane&15            ; frag.x = K=2*hi, frag.y = K=2*hi+1
//   C 16x16: lane -> col N=lane&15 ; VGPR r -> row M = r + 8*hi
// ---------------------------------------------------------------------------
__global__ __launch_bounds__(32) void wmma_gemm_f32(
    const float* __restrict__ A, const float* __restrict__ W,
    float* __restrict__ C, int N) {
  const int lane = threadIdx.x & 31;
  const int li = lane & 15;
  const int hi = lane >> 4;
  const int n0 = blockIdx.x * 16;
  const int m0 = blockIdx.y * 16;

  const float* arow = A + (size_t)(m0 + li) * HID;
  const float* wcol = W + n0 + li;

  v8f acc = {};
  for (int k = 0; k < HID; k += 4) {
    v2f a = *(const v2f*)(arow + k + 2 * hi);
    v2f b;
    b.x = wcol[(size_t)(k + 2 * hi) * N];
    b.y = wcol[(size_t)(k + 2 * hi + 1) * N];
    acc = __builtin_amdgcn_wmma_f32_16x16x4_f32(false, a, false, b,
                                                (short)0, acc, false, false);
  }
  float* crow = C + (size_t)(m0 + 8 * hi) * N + n0 + li;
#pragma unroll
  for (int r = 0; r < 8; ++r) crow[(size_t)r * N] = acc[r];
}

// ---------------------------------------------------------------------------
// RoPE on Q (32 heads) and K, extract V. One block per batch row.
// qkv layout: [b][ H*D (q) | D (k) | D (v) ]
// ---------------------------------------------------------------------------
__global__ __launch_bounds__(128) void rope_scatter(
    const float* __restrict__ qkv, const int* __restrict__ positions,
    float* __restrict__ qrot, float* __restrict__ krot,
    float* __restrict__ vnew) {
  const int b = blockIdx.x;
  const int tid = threadIdx.x;
  __shared__ float cs[64], sn[64];

  const float pos = (float)positions[b];
  for (int i = tid; i < 64; i += 128) {
    float inv = powf(10000.0f, -(float)(2 * i) / 128.0f);
    float ang = pos * inv;
    cs[i] = cosf(ang);
    sn[i] = sinf(ang);
  }
  __syncthreads();

  const float* src = qkv + (size_t)b * QKV_N;
  // Q heads
  for (int idx = tid; idx < NHEAD * 64; idx += 128) {
    int h = idx >> 6, i = idx & 63;
    float x1 = src[h * HD + i];
    float x2 = src[h * HD + 64 + i];
    float* dst = qrot + (size_t)b * (NHEAD * HD) + h * HD;
    dst[i]      = x1 * cs[i] - x2 * sn[i];
    dst[64 + i] = x2 * cs[i] + x1 * sn[i];
  }
  // K
  for (int i = tid; i < 64; i += 128) {
    float x1 = src[NHEAD * HD + i];
    float x2 = src[NHEAD * HD + 64 + i];
    krot[(size_t)b * HD + i]      = x1 * cs[i] - x2 * sn[i];
    krot[(size_t)b * HD + 64 + i] = x2 * cs[i] + x1 * sn[i];
  }
  // V
  for (int i = tid; i < HD; i += 128)
    vnew[(size_t)b * HD + i] = src[(NHEAD + 1) * HD + i];
}

// ---------------------------------------------------------------------------
// Paged flash-decode attention, fp32 WMMA.
// One workgroup (4 waves) per batch; waves stride over 16-key tiles.
// New-token K/V substituted at l == pos (caches are read-only inputs).
// ---------------------------------------------------------------------------
__global__ __launch_bounds__(128) void paged_attn_wmma(
    const float* __restrict__ Q,    // B x 32 x 128 (rope'd)
    const float* __restrict__ Knew, // B x 128
    const float* __restrict__ Vnew, // B x 128
    const float* __restrict__ Kc,   // nblk x 16 x 128
    const float* __restrict__ Vc,
    const int* __restrict__ positions,
    const int* __restrict__ block_tables, // B x 256
    const int* __restrict__ ctx_lens,
    float* __restrict__ O) {        // B x 32 x 128
  const int b = blockIdx.x;
  const int tid = threadIdx.x;
  const int w = tid >> 5;
  const int lane = tid & 31;
  const int li = lane & 15;
  const int hi = lane >> 4;

  __shared__ float sm_p[4][32 * 16];  // per-wave P-tile staging (layout shuffle)
  __shared__ float sm_m[4][32];
  __shared__ float sm_l[4][32];
  __shared__ float sm_o[32 * 128];

  for (int i = tid; i < 32 * 128; i += 128) sm_o[i] = 0.0f;
  __syncthreads();

  const int ctx = ctx_lens[b];
  const int pos = positions[b];
  const int T = (ctx + 15) >> 4;
  const float* Qb = Q + (size_t)b * (NHEAD * HD);
  const float scale = 0.08838834764831845f; // 1/sqrt(128)

  v8f acc[2][8];
  float rm[2][8], rl[2][8];
#pragma unroll
  for (int mt = 0; mt < 2; ++mt)
#pragma unroll
    for (int r = 0; r < 8; ++r) { rm[mt][r] = NEGINF; rl[mt][r] = 0.f; }
#pragma unroll
  for (int mt = 0; mt < 2; ++mt)
#pragma unroll
    for (int nt = 0; nt < 8; ++nt) acc[mt][nt] = (v8f){};

  for (int t = w; t < T; t += 4) {
    const int blk = block_tables[b * MAXBLK + t];
    const int kbase = t * BLKSZ;

    // ---- scores S(32x16) = Q(32x128) x K^T(128x16) : 64 WMMAs ----
    const int kj = kbase + li; // key index of this lane's score column
    const float* krow = (kj == pos) ? (Knew + (size_t)b * HD)
                                    : (Kc + ((size_t)blk * BLKSZ + li) * HD);
    v8f s[2];
    s[0] = (v8f){}; s[1] = (v8f){};
    for (int kk = 0; kk < HD; kk += 4) {
      v2f bfrag = *(const v2f*)(krow + kk + 2 * hi);
#pragma unroll
      for (int mt = 0; mt < 2; ++mt) {
        v2f afrag = *(const v2f*)(Qb + (size_t)(mt * 16 + li) * HD + kk + 2 * hi);
        s[mt] = __builtin_amdgcn_wmma_f32_16x16x4_f32(
            false, afrag, false, bfrag, (short)0, s[mt], false, false);
      }
    }

    // ---- online softmax (rows=heads spread over 8 VGPRs, cols=keys over lanes) ----
    const bool valid = (kbase + li) < ctx;
#pragma unroll
    for (int mt = 0; mt < 2; ++mt) {
      float p[8], tm[8], ts[8], alpha[8];
#pragma unroll
      for (int r = 0; r < 8; ++r) {
        p[r] = valid ? s[mt][r] * scale : NEGINF;
        tm[r] = p[r];
      }
      // row max over the 16 key-columns of this lane half
#pragma unroll
      for (int r = 0; r < 8; ++r) {
#pragma unroll
        for (int msk = 1; msk <= 8; msk <<= 1)
          tm[r] = fmaxf(tm[r], __shfl_xor(tm[r], msk, 32));
      }
#pragma unroll
      for (int r = 0; r < 8; ++r) {
        float nm = fmaxf(rm[mt][r], tm[r]);
        alpha[r] = expf(rm[mt][r] - nm);
        p[r] = expf(p[r] - nm);
        ts[r] = p[r];
        rm[mt][r] = nm;
      }
#pragma unroll
      for (int r = 0; r < 8; ++r) {
#pragma unroll
        for (int msk = 1; msk <= 8; msk <<= 1)
          ts[r] += __shfl_xor(ts[r], msk, 32);
        rl[mt][r] = rl[mt][r] * alpha[r] + ts[r];
      }
#pragma unroll
      for (int nt = 0; nt < 8; ++nt)
#pragma unroll
        for (int r = 0; r < 8; ++r) acc[mt][nt][r] *= alpha[r];
      // stage P into LDS in (row-major 32x16) so it can be re-read as A frags
#pragma unroll
      for (int r = 0; r < 8; ++r)
        sm_p[w][(mt * 16 + r + 8 * hi) * 16 + li] = p[r];
    }

    // ---- acc += P(32x16) x V(16x128) : 64 WMMAs ----
    // (LDS RAW within a wave is safe: DS ops are in-order per wave)
#pragma unroll
    for (int kk = 0; kk < 16; kk += 4) {
      const int kv0 = kbase + kk + 2 * hi;
      const float* vr0 = (kv0 == pos)
                             ? (Vnew + (size_t)b * HD)
                             : (Vc + ((size_t)blk * BLKSZ + (kk + 2 * hi)) * HD);
      const float* vr1 = ((kv0 + 1) == pos)
                             ? (Vnew + (size_t)b * HD)
                             : (Vc + ((size_t)blk * BLKSZ + (kk + 2 * hi + 1)) * HD);
#pragma unroll
      for (int mt = 0; mt < 2; ++mt) {
        v2f afrag = *(const v2f*)(&sm_p[w][(mt * 16 + li) * 16 + kk + 2 * hi]);
#pragma unroll
        for (int nt = 0; nt < 8; ++nt) {
          v2f bfrag;
          bfrag.x = vr0[nt * 16 + li];
          bfrag.y = vr1[nt * 16 + li];
          acc[mt][nt] = __builtin_amdgcn_wmma_f32_16x16x4_f32(
              false, afrag, false, bfrag, (short)0, acc[mt][nt], false, false);
        }
      }
    }
  }

  // ---- cross-wave combine ----
  if (li == 0) {
#pragma unroll
    for (int mt = 0; mt < 2; ++mt)
#pragma unroll
      for (int r = 0; r < 8; ++r) {
        int row = mt * 16 + r + 8 * hi;
        sm_m[w][row] = rm[mt][r];
        sm_l[w][row] = rl[mt][r];
      }
  }
  __syncthreads();

#pragma unroll
  for (int mt = 0; mt < 2; ++mt) {
#pragma unroll
    for (int r = 0; r < 8; ++r) {
      int row = mt * 16 + r + 8 * hi;
      float mg = fmaxf(fmaxf(sm_m[0][row], sm_m[1][row]),
                       fmaxf(sm_m[2][row], sm_m[3][row]));
      float f = expf(rm[mt][r] - mg);
#pragma unroll
      for (int nt = 0; nt < 8; ++nt)
        atomicAdd(&sm_o[row * 128 + nt * 16 + li], acc[mt][nt][r] * f);
    }
  }
  __syncthreads();

  float* Ob = O + (size_t)b * (NHEAD * HD);
  for (int i = tid; i < NHEAD * HD; i += 128) {
    int row = i >> 7;
    float mg = fmaxf(fmaxf(sm_m[0][row], sm_m[1][row]),
                     fmaxf(sm_m[2][row], sm_m[3][row]));
    float lg = 0.f;
#pragma unroll
    for (int ww = 0; ww < 4; ++ww)
      lg += sm_l[ww][row] * expf(sm_m[ww][row] - mg);
    Ob[i] = sm_o[i] / lg;
  }
}

// ---------------------------------------------------------------------------
extern "C" void kernel_launch(void* const* d_in, const int* in_sizes, int n_in,
                              void* d_out, int out_size, void* d_ws,
                              size_t ws_size, hipStream_t stream) {
  const float* hs     = (const float*)d_in[0]; // 64 x 4096
  const float* wqkv   = (const float*)d_in[1]; // 4096 x 4352
  const float* wdense = (const float*)d_in[2]; // 4096 x 4096
  const float* kc     = (const float*)d_in[3]; // 16384 x 16 x 128
  const float* vc     = (const float*)d_in[4];
  const int* positions = (const int*)d_in[5];  // B
  const int* btab      = (const int*)d_in[6];  // B x 256
  const int* ctx       = (const int*)d_in[7];  // B

  float* ws   = (float*)d_ws;
  float* qkv  = ws;                               // 64*4352
  float* qrot = qkv + (size_t)BATCH * QKV_N;      // 64*4096
  float* krot = qrot + (size_t)BATCH * NHEAD * HD;// 64*128
  float* vnew = krot + (size_t)BATCH * HD;        // 64*128
  float* attn = vnew + (size_t)BATCH * HD;        // 64*4096

  // 1) qkv = hs @ wqkv
  wmma_gemm_f32<<<dim3(QKV_N / 16, BATCH / 16), 32, 0, stream>>>(
      hs, wqkv, qkv, QKV_N);
  // 2) RoPE on q,k; extract v
  rope_scatter<<<BATCH, 128, 0, stream>>>(qkv, positions, qrot, krot, vnew);
  // 3) paged attention
  paged_attn_wmma<<<BATCH, 128, 0, stream>>>(qrot, krot, vnew, kc, vc,
                                             positions, btab, ctx, attn);
  // 4) out = attn @ wdense
  wmma_gemm_f32<<<dim3(HID / 16, BATCH / 16), 32, 0, stream>>>(
      attn, wdense, (float*)d_out, HID);
}